// SPDconv_66331474919501
// MI455X (gfx1250) — compile-verified
//
#include <hip/hip_runtime.h>
#include <math.h>

typedef __attribute__((ext_vector_type(2))) float v2f;
typedef __attribute__((ext_vector_type(8))) float v8f;

#define IMG    256
#define KP     256      // 16*16 kernel positions
#define LP     1089     // 33*33 patch positions
#define LPAD   1152     // 18*64
#define M2     2178
#define M2PAD  2240     // 35*64
#define EPSV   1e-4f

#define SIGN_ITERS   14
#define SQRT_LEVELS  5
#define SQRT_ITERS   9
#define LOG_TERMS    12

// ---------------------------------------------------------------------------
// WMMA fp32 GEMM: C = alpha * op(A) * op(B)
// TA=1: A stored K x M (we use A^T).  TB=1: B stored N x K (we use B^T).
// One wave computes a 16(M) x 64(N) tile of C: 4 accumulators share each
// A fragment (4x less A traffic, 4 independent WMMA chains per K-step).
// All M dims are multiples of 16 and N dims multiples of 64; EXEC all ones.
// A-frag layout (16x4 f32): lane 0-15 -> M=lane, K = 0(v0)/1(v1);
//                           lane 16-31 -> M=lane-16, K = 2(v0)/3(v1).
// C layout: VGPR r -> row m0+r (lanes 0-15) / m0+8+r (lanes 16-31), col=lane&15.
// ---------------------------------------------------------------------------
template<int TA, int TB>
__global__ __launch_bounds__(32)
void wmma_gemm(const float* __restrict__ A, const float* __restrict__ B,
               float* __restrict__ C, int K, int lda, int ldb, int ldc,
               float alpha)
{
  const int lane = threadIdx.x;
  const int m0   = blockIdx.y * 16;
  const int n0   = blockIdx.x * 64;
  const int l15  = lane & 15;
  const int koff = (lane >> 4) * 2;   // 0 or 2
  v8f acc[4] = {};
#pragma unroll 2
  for (int k = 0; k < K; k += 4) {
    const int ka = k + koff;
    v2f a;
    if (TA) { a.x = A[(size_t)ka * lda + m0 + l15];
              a.y = A[(size_t)(ka + 1) * lda + m0 + l15]; }
    else    { a.x = A[(size_t)(m0 + l15) * lda + ka];
              a.y = A[(size_t)(m0 + l15) * lda + ka + 1]; }
#pragma unroll
    for (int j = 0; j < 4; ++j) {
      v2f b;
      const int nj = n0 + 16 * j + l15;
      if (TB) { b.x = B[(size_t)nj * ldb + ka];
                b.y = B[(size_t)nj * ldb + ka + 1]; }
      else    { b.x = B[(size_t)ka * ldb + nj];
                b.y = B[(size_t)(ka + 1) * ldb + nj]; }
      acc[j] = __builtin_amdgcn_wmma_f32_16x16x4_f32(false, a, false, b,
                                                     (short)0, acc[j],
                                                     false, false);
    }
  }
  const int mrow = m0 + ((lane >> 4) << 3);
#pragma unroll
  for (int j = 0; j < 4; ++j)
#pragma unroll
    for (int r = 0; r < 8; ++r)
      C[(size_t)(mrow + r) * ldc + n0 + 16 * j + l15] = alpha * acc[j][r];
}

// ---------------------------------------------------------------------------
// Elementwise / setup kernels
// ---------------------------------------------------------------------------
__global__ void k_unfold(const float* __restrict__ img, float* __restrict__ P)
{
  int idx = blockIdx.x * 256 + threadIdx.x;
  if (idx >= LPAD * KP) return;
  int l = idx >> 8, c = idx & 255;
  float v = 0.0f;
  if (l < LP) {
    int py = l / 33, px = l - py * 33;
    int ky = c >> 4, kx = c & 15;
    int y = py * 8 - 8 + ky, x = px * 8 - 8 + kx;
    if ((unsigned)y < IMG && (unsigned)x < IMG) v = img[y * IMG + x];
  }
  P[idx] = v;
}

__global__ void k_center(const float* __restrict__ Pir,
                         const float* __restrict__ Pvi,
                         float* __restrict__ Xc)
{
  __shared__ float red[256];
  int r = blockIdx.x, c = threadIdx.x;
  float v = 0.0f;
  if (r < M2) v = (r < LP) ? Pir[r * KP + c] : Pvi[(r - LP) * KP + c];
  red[c] = v;
  __syncthreads();
  for (int s = 128; s > 0; s >>= 1) {
    if (c < s) red[c] += red[c + s];
    __syncthreads();
  }
  float mean = red[0] * (1.0f / (float)KP);
  Xc[(size_t)r * KP + c] = (r < M2) ? (v - mean) : 0.0f;
}

__global__ void k_padW(const float* __restrict__ W, float* __restrict__ Wp)
{
  int idx = blockIdx.x * 256 + threadIdx.x;
  if (idx >= M2PAD * LPAD) return;
  int r = idx / LPAD, c = idx - r * LPAD;
  Wp[idx] = (r < M2 && c < LP) ? W[(size_t)r * LP + c] : 0.0f;
}

__global__ void k_axpby(float* __restrict__ C, const float* __restrict__ A,
                        const float* __restrict__ B, float a, float b, int n)
{
  int i = blockIdx.x * 256 + threadIdx.x;
  if (i < n) C[i] = a * A[i] + b * B[i];
}

__global__ void k_zero(float* __restrict__ A, int n)
{
  int i = blockIdx.x * 256 + threadIdx.x;
  if (i < n) A[i] = 0.0f;
}

__global__ void k_adddiag(float* __restrict__ A, float val, int n, int ld)
{
  int i = blockIdx.x * 256 + threadIdx.x;
  if (i < n) A[(size_t)i * ld + i] += val;
}

__global__ void k_adddiag_log(float* __restrict__ A, const float* __restrict__ s,
                              int n, int ld)
{
  int i = blockIdx.x * 256 + threadIdx.x;
  if (i < n) A[(size_t)i * ld + i] += logf(*s);
}

__global__ void k_cov_diag(float* __restrict__ cov, const float* __restrict__ trp,
                           int n, int ld)
{
  int i = blockIdx.x * 256 + threadIdx.x;
  if (i < n) {
    float c = 0.001f * (*trp);
    cov[(size_t)i * ld + i] += c * (1.0f + 1.0e-3f * ((float)i / (float)n));
  }
}

__global__ void k_scale_rsqrt(float* __restrict__ Y, const float* __restrict__ X,
                              const float* __restrict__ s, int n)
{
  int i = blockIdx.x * 256 + threadIdx.x;
  if (i < n) Y[i] = X[i] * rsqrtf(*s);
}

__global__ void k_scale_inv(float* __restrict__ Y, const float* __restrict__ X,
                            const float* __restrict__ s, int n)
{
  int i = blockIdx.x * 256 + threadIdx.x;
  if (i < n) Y[i] = X[i] / (*s);
}

__global__ void k_trace(const float* __restrict__ A, int n, int ld,
                        float* __restrict__ out)
{
  __shared__ float red[256];
  float s = 0.0f;
  for (int i = threadIdx.x; i < n; i += 256) s += A[(size_t)i * ld + i];
  red[threadIdx.x] = s;
  __syncthreads();
  for (int st = 128; st > 0; st >>= 1) {
    if (threadIdx.x < st) red[threadIdx.x] += red[threadIdx.x + st];
    __syncthreads();
  }
  if (threadIdx.x == 0) *out = red[0];
}

__global__ void k_sumsq(const float* __restrict__ A, int n,
                        float* __restrict__ out)
{
  __shared__ float red[256];
  float s = 0.0f;
  for (int i = threadIdx.x; i < n; i += 256) { float v = A[i]; s += v * v; }
  red[threadIdx.x] = s;
  __syncthreads();
  for (int st = 128; st > 0; st >>= 1) {
    if (threadIdx.x < st) red[threadIdx.x] += red[threadIdx.x + st];
    __syncthreads();
  }
  if (threadIdx.x == 0) *out = red[0];
}

__global__ void k_strip(float* __restrict__ dst, const float* __restrict__ src,
                        int n, int lds)
{
  long long i = (long long)blockIdx.x * 256 + threadIdx.x;
  long long total = (long long)n * n;
  if (i < total) {
    int r = (int)(i / n), c = (int)(i - (long long)r * n);
    dst[i] = src[(size_t)r * lds + c];
  }
}

// fold (transpose of unfold): gather the <=4 patches covering each pixel
__global__ void k_fold(const float* __restrict__ O, float* __restrict__ res)
{
  int idx = blockIdx.x * 256 + threadIdx.x;
  if (idx >= IMG * IMG) return;
  int y = idx >> 8, x = idx & 255;
  float s = 0.0f;
  for (int ky = 0; ky < 16; ++ky) {
    int ty = y + 8 - ky;
    if (ty & 7) continue;
    int py = ty >> 3;
    if (py < 0 || py >= 33) continue;
    for (int kx = 0; kx < 16; ++kx) {
      int tx = x + 8 - kx;
      if (tx & 7) continue;
      int px = tx >> 3;
      if (px < 0 || px >= 33) continue;
      s += O[(size_t)(py * 33 + px) * KP + (ky * 16 + kx)];
    }
  }
  res[idx] = s;
}

// ---------------------------------------------------------------------------
extern "C" void kernel_launch(void* const* d_in, const int* in_sizes, int n_in,
                              void* d_out, int out_size, void* d_ws, size_t ws_size,
                              hipStream_t stream)
{
  (void)in_sizes; (void)n_in; (void)out_size; (void)ws_size;
  const float* ir = (const float*)d_in[0];
  const float* vi = (const float*)d_in[1];
  const float* W  = (const float*)d_in[2];
  float* out = (float*)d_out;
  float* ws  = (float*)d_ws;

  const size_t LPKP  = (size_t)LPAD * KP;      // 294912
  const size_t M2KP  = (size_t)M2PAD * KP;     // 573440
  const size_t WSZ   = (size_t)M2PAD * LPAD;   // 2580480
  const size_t COVSZ = (size_t)M2PAD * M2PAD;  // 5017600
  const size_t NSZ   = (size_t)LPAD * LPAD;    // 1327104
  const int    NSZi  = (int)NSZ;

  float* scal = ws;                 // [64] scalars
  float* Pir  = ws + 64;
  float* Pvi  = Pir + LPKP;
  float* Xc   = Pvi + LPKP;
  float* Wp   = Xc + M2KP;
  float* covp = Wp + WSZ;
  float* Tp   = covp + COVSZ;
  float* X1p  = Tp + WSZ;
  float* Bm   = X1p + NSZ;
  float* G0   = Bm + NSZ;
  float* G1   = G0 + NSZ;
  float* G2   = G1 + NSZ;
  float* G3   = G2 + NSZ;
  float* G4   = G3 + NSZ;
  float* G5   = G4 + NSZ;
  float* X2p  = G5 + NSZ;
  float* X3p  = X2p + NSZ;
  float* O1   = X3p + NSZ;
  float* O2   = O1 + LPKP;

  float* o_r1 = out;
  float* o_r2 = o_r1 + IMG * IMG;
  float* o_s1 = o_r2 + IMG * IMG;
  float* o_s2 = o_s1 + (size_t)M2 * M2;
  float* o_x2 = o_s2 + (size_t)M2 * M2;
  float* o_x3 = o_x2 + (size_t)LP * LP;

#define EW(kern, n, ...) \
  kern<<<dim3((unsigned)(((size_t)(n) + 255) / 256)), dim3(256), 0, stream>>>(__VA_ARGS__)
#define GEMM(TA, TB, A_, B_, C_, Mt, Nt, Kt, lda, ldb, ldc, al) \
  wmma_gemm<TA, TB><<<dim3((Nt) / 64, (Mt) / 16), dim3(32), 0, stream>>>( \
      A_, B_, C_, Kt, lda, ldb, ldc, al)

  // 1) unfold both images into zero-padded patch matrices [LPAD x 256]
  EW(k_unfold, LPKP, ir, Pir);
  EW(k_unfold, LPKP, vi, Pvi);

  // 2) row-centered X  [M2PAD x 256] (pad rows = 0)
  k_center<<<dim3(M2PAD), dim3(256), 0, stream>>>(Pir, Pvi, Xc);

  // 3) zero-padded W  [M2PAD x LPAD]
  EW(k_padW, WSZ, W, Wp);

  // 4) cov = Xc Xc^T / 255   (fp32 WMMA, NT)
  GEMM(0, 1, Xc, Xc, covp, M2PAD, M2PAD, KP, KP, KP, M2PAD, 1.0f / 255.0f);

  // 5) diagonal regularization: c = 1e-3*tr ; diag += c*(1 + 1e-3*i/n)
  k_trace<<<dim3(1), dim3(256), 0, stream>>>(covp, M2, M2PAD, scal + 0);
  EW(k_cov_diag, M2, covp, scal + 0, M2, M2PAD);

  // 6) spd1 outputs (twice)
  EW(k_strip, (size_t)M2 * M2, o_s1, covp, M2, M2PAD);
  EW(k_strip, (size_t)M2 * M2, o_s2, covp, M2, M2PAD);

  // 7) BiMap: T = cov W ; X1 = W^T T
  GEMM(0, 0, covp, Wp, Tp, M2PAD, LPAD, M2PAD, M2PAD, LPAD, LPAD, 1.0f);
  GEMM(1, 0, Wp, Tp, X1p, LPAD, LPAD, M2PAD, LPAD, LPAD, LPAD, 1.0f);

  // 8) ReEig: X2 = (X1 + eps I + |X1 - eps I|)/2 ; |B| = B sign(B),
  //    sign via Newton-Schulz S <- 0.5 S (3I - S^2)
  EW(k_axpby, NSZ, Bm, X1p, X1p, 1.0f, 0.0f, NSZi);
  EW(k_adddiag, LP, Bm, -EPSV, LP, LPAD);
  k_sumsq<<<dim3(1), dim3(256), 0, stream>>>(Bm, NSZi, scal + 1);
  {
    float *S = G0, *S2 = G1, *E = G2, *Sn = G3;
    EW(k_scale_rsqrt, NSZ, S, Bm, scal + 1, NSZi);
    for (int it = 0; it < SIGN_ITERS; ++it) {
      GEMM(0, 0, S, S, S2, LPAD, LPAD, LPAD, LPAD, LPAD, LPAD, 1.0f);
      EW(k_axpby, NSZ, E, S2, S2, -0.5f, 0.0f, NSZi);
      EW(k_adddiag, LP, E, 1.5f, LP, LPAD);
      GEMM(0, 0, S, E, Sn, LPAD, LPAD, LPAD, LPAD, LPAD, LPAD, 1.0f);
      float* t = S; S = Sn; Sn = t;
    }
    float* AbsB = S2;
    GEMM(0, 0, Bm, S, AbsB, LPAD, LPAD, LPAD, LPAD, LPAD, LPAD, 1.0f);
    EW(k_axpby, NSZ, X2p, X1p, AbsB, 0.5f, 0.5f, NSZi);
    EW(k_adddiag, LP, X2p, 0.5f * EPSV, LP, LPAD);
  }
  EW(k_strip, (size_t)LP * LP, o_x2, X2p, LP, LPAD);

  // 9) LogEig: A0 = X2/tr (eigs in (0,1]); k repeated sqrt via coupled
  //    Newton-Schulz; Mercator series near I; X3 = 2^k log(A_k) + log(tr) I
  k_trace<<<dim3(1), dim3(256), 0, stream>>>(X2p, LP, LPAD, scal + 2);
  float *Y = G0, *Z = G1, *Wm = G2, *Ee = G3, *Yn = G4, *Zn = G5;
  EW(k_scale_inv, NSZ, Y, X2p, scal + 2, NSZi);
  EW(k_zero, NSZ, Z, NSZi);
  EW(k_adddiag, LP, Z, 1.0f, LP, LPAD);
  for (int lvl = 0; lvl < SQRT_LEVELS; ++lvl) {
    for (int it = 0; it < SQRT_ITERS; ++it) {
      GEMM(0, 0, Z, Y, Wm, LPAD, LPAD, LPAD, LPAD, LPAD, LPAD, 1.0f);
      EW(k_axpby, NSZ, Ee, Wm, Wm, -0.5f, 0.0f, NSZi);
      EW(k_adddiag, LP, Ee, 1.5f, LP, LPAD);
      GEMM(0, 0, Y, Ee, Yn, LPAD, LPAD, LPAD, LPAD, LPAD, LPAD, 1.0f);
      GEMM(0, 0, Ee, Z, Zn, LPAD, LPAD, LPAD, LPAD, LPAD, LPAD, 1.0f);
      float* t;
      t = Y; Y = Yn; Yn = t;
      t = Z; Z = Zn; Zn = t;
    }
    if (lvl + 1 < SQRT_LEVELS) {       // reset Z = I for next sqrt level
      EW(k_zero, NSZ, Z, NSZi);
      EW(k_adddiag, LP, Z, 1.0f, LP, LPAD);
    }
  }
  {
    // M = A_k - I ; Horner: log ~= M(c1 I + M(c2 I + ... c_T I))
    float* Mm = Wm;                    // G2, Y not aliased
    EW(k_axpby, NSZ, Mm, Y, Y, 1.0f, 0.0f, NSZi);
    EW(k_adddiag, LP, Mm, -1.0f, LP, LPAD);
    float* R  = Ee;                    // G3
    float* Tm = Bm;                    // Bm free now
    EW(k_zero, NSZ, R, NSZi);
    float cT = ((LOG_TERMS & 1) ? 1.0f : -1.0f) / (float)LOG_TERMS;
    EW(k_adddiag, LP, R, cT, LP, LPAD);
    for (int t = LOG_TERMS - 1; t >= 1; --t) {
      GEMM(0, 0, Mm, R, Tm, LPAD, LPAD, LPAD, LPAD, LPAD, LPAD, 1.0f);
      float ct = ((t & 1) ? 1.0f : -1.0f) / (float)t;
      EW(k_adddiag, LP, Tm, ct, LP, LPAD);
      float* tmp = R; R = Tm; Tm = tmp;
    }
    float* Pl = X1p;                   // X1p free now
    GEMM(0, 0, Mm, R, Pl, LPAD, LPAD, LPAD, LPAD, LPAD, LPAD, 1.0f);
    EW(k_axpby, NSZ, X3p, Pl, Pl, (float)(1 << SQRT_LEVELS), 0.0f, NSZi);
    EW(k_adddiag_log, LP, X3p, scal + 2, LP, LPAD);
  }
  EW(k_strip, (size_t)LP * LP, o_x3, X3p, LP, LPAD);

  // 10) out1 = X3 pir ; out2 = X3 pvi ; fold back to images
  GEMM(0, 0, X3p, Pir, O1, LPAD, KP, LPAD, LPAD, KP, KP, 1.0f);
  GEMM(0, 0, X3p, Pvi, O2, LPAD, KP, LPAD, LPAD, KP, KP, 1.0f);
  EW(k_fold, IMG * IMG, O1, o_r1);
  EW(k_fold, IMG * IMG, O2, o_r2);

#undef GEMM
#undef EW
}